// UzmanKapisi_30030411334250
// MI455X (gfx1250) — compile-verified
//
#include <hip/hip_runtime.h>
#include <hip/hip_bf16.h>
#include <math.h>

// gfx1250 is wave32; WMMA f32 16x16x4: A=16x4 (2 VGPR), B=4x16 (2 VGPR), C/D=16x16 (8 VGPR)
typedef __attribute__((ext_vector_type(2))) float v2f;
typedef __attribute__((ext_vector_type(8))) float v8f;

#define LOKAL_GRUP 7
#define TOPK 2
#define EPS_F 1e-8f
#define CACHE_BONUS_F 0.1f

#define D_DIM 1024
#define E_NUM 64
#define N_TOK 32768           // B*S = 8*4096
#define WG_THREADS 128        // 4 waves of 32
#define TOK_PER_WAVE 32       // 2 B-tiles of 16 tokens per wave
#define TOK_PER_WG 128
#define SIM_STRIDE 65         // odd stride -> bank(token row) = token mod 64, conflict-free

__global__ __launch_bounds__(WG_THREADS)
void uzman_kapisi_kernel(const float* __restrict__ tokp,
                         const float* __restrict__ expp,
                         const float* __restrict__ cachep,
                         int* __restrict__ outIdx,
                         float* __restrict__ outW)
{
    __shared__ float simLds[TOK_PER_WG * SIM_STRIDE];   // 33280 B
    __shared__ float normPart[TOK_PER_WG * 2];          // per-token sumsq halves
    __shared__ float expSumSq[E_NUM];
    __shared__ float expInv[E_NUM];
    __shared__ float bonus[E_NUM];

    const int tid  = threadIdx.x;
    const int lane = tid & 31;
    const int wave = tid >> 5;
    const int wgStart = blockIdx.x * TOK_PER_WG;

    // ---------- expert sum-of-squares + cache bonus (L2-hot, duplicated per WG) ----------
    if (tid < E_NUM) expSumSq[tid] = 0.0f;
    __syncthreads();
    {
        const int e = tid & (E_NUM - 1);
        const int half = tid >> 6;                       // 0 or 1
        const float* row = expp + e * D_DIM + half * (D_DIM / 2);
        float s = 0.0f;
        #pragma unroll 4
        for (int i = 0; i < D_DIM / 2; ++i) s += row[i] * row[i];
        atomicAdd(&expSumSq[e], s);                      // ds_add_f32
    }
    __syncthreads();
    if (tid < E_NUM) {
        expInv[tid] = 1.0f / (sqrtf(expSumSq[tid]) + EPS_F);
        bonus[tid]  = CACHE_BONUS_F * cachep[tid];
    }
    // visibility to all waves is guaranteed by the barrier after the sim dump

    // ---------- fp32 WMMA GEMM: sim(e, t) = experts(64 x 1024) x tokens^T ----------
    // A tile (16x4):  lane l -> M = 16m + (l%16), K = k0 + 2*(l/16) + {0,1}
    // B tile (4x16):  lane l -> N = token, K = k0 + 2*(l/16) + {0,1}
    const int mrow  = lane & 15;
    const int khalf = (lane >> 4) * 2;
    const int tBase = wgStart + wave * TOK_PER_WAVE;

    const float* aPtr[4];
    const float* bPtr[2];
    #pragma unroll
    for (int m = 0; m < 4; ++m)
        aPtr[m] = expp + (m * 16 + mrow) * D_DIM + khalf;
    #pragma unroll
    for (int n = 0; n < 2; ++n)
        bPtr[n] = tokp + (size_t)(tBase + n * 16 + mrow) * D_DIM + khalf;

    v8f c[4][2];
    #pragma unroll
    for (int m = 0; m < 4; ++m)
        #pragma unroll
        for (int n = 0; n < 2; ++n)
            c[m][n] = (v8f)(0.0f);

    float sq0 = 0.0f, sq1 = 0.0f;   // per-lane token sumsq partials (half of dims each)

    #pragma unroll 2
    for (int k0 = 0; k0 < D_DIM; k0 += 4) {
        v2f a[4], b[2];
        #pragma unroll
        for (int m = 0; m < 4; ++m) a[m] = *(const v2f*)(aPtr[m] + k0);   // global_load_b64
        b[0] = *(const v2f*)(bPtr[0] + k0);
        b[1] = *(const v2f*)(bPtr[1] + k0);
        sq0 += b[0].x * b[0].x + b[0].y * b[0].y;
        sq1 += b[1].x * b[1].x + b[1].y * b[1].y;
        #pragma unroll
        for (int m = 0; m < 4; ++m) {
            c[m][0] = __builtin_amdgcn_wmma_f32_16x16x4_f32(
                false, a[m], false, b[0], (short)0, c[m][0], false, false);
            c[m][1] = __builtin_amdgcn_wmma_f32_16x16x4_f32(
                false, a[m], false, b[1], (short)0, c[m][1], false, false);
        }
    }

    // ---------- transpose sims through LDS (D layout: VGPR r -> M=16m+r+8*(l/16), N=l%16) ----------
    #pragma unroll
    for (int n = 0; n < 2; ++n) {
        const int tokLocal = wave * TOK_PER_WAVE + n * 16 + mrow;
        normPart[tokLocal * 2 + (lane >> 4)] = (n == 0) ? sq0 : sq1;
        #pragma unroll
        for (int m = 0; m < 4; ++m) {
            #pragma unroll
            for (int r = 0; r < 8; ++r) {
                const int col = m * 16 + r + 8 * (lane >> 4);
                simLds[tokLocal * SIM_STRIDE + col] = c[m][n][r];
            }
        }
    }
    __syncthreads();

    // ---------- per-token routing: one thread = one token ----------
    const int t = tid;
    const float* srow = &simLds[t * SIM_STRIDE];
    const float invT = 1.0f / (sqrtf(normPart[2 * t] + normPart[2 * t + 1]) + EPS_F);

    float simv[E_NUM];
    #pragma unroll 8
    for (int e = 0; e < E_NUM; ++e) simv[e] = srow[e];

    // top-7 of raw sim, descending, first-index tie-break (matches lax.top_k)
    int descIdx[LOKAL_GRUP];
    #pragma unroll
    for (int j = 0; j < LOKAL_GRUP; ++j) {
        float best = -__builtin_inff();
        int bi = 0;
        for (int e = 0; e < E_NUM; ++e) {
            if (simv[e] > best) { best = simv[e]; bi = e; }
        }
        descIdx[j] = bi;
        simv[bi] = -__builtin_inff();
    }

    // reversed -> ascending local_idx; total = cos + bonus, where
    // cos = sim * invT * invE  (exact: normalization is elementwise scaling)
    int   localIdx[LOKAL_GRUP];
    float tot[LOKAL_GRUP];
    #pragma unroll
    for (int j = 0; j < LOKAL_GRUP; ++j) {
        const int li = descIdx[LOKAL_GRUP - 1 - j];
        localIdx[j] = li;
        tot[j] = srow[li] * invT * expInv[li] + bonus[li];
    }

    // top-2 of tot (descending, first-index tie-break), then reversed
    int p0 = 0; float v0 = tot[0];
    #pragma unroll
    for (int j = 1; j < LOKAL_GRUP; ++j)
        if (tot[j] > v0) { v0 = tot[j]; p0 = j; }
    int p1 = 0; float v1 = -__builtin_inff();
    #pragma unroll
    for (int j = 0; j < LOKAL_GRUP; ++j)
        if (j != p0 && tot[j] > v1) { v1 = tot[j]; p1 = j; }

    // softmax over [v1, v0] (ascending order after [::-1])
    const float e1 = __expf(v1 - v0);      // v0 is the max
    const float inv = 1.0f / (1.0f + e1);

    const int g = wgStart + t;
    outIdx[g * TOPK + 0] = localIdx[p1];
    outIdx[g * TOPK + 1] = localIdx[p0];
    outW[g * TOPK + 0] = e1 * inv;
    outW[g * TOPK + 1] = inv;
}

extern "C" void kernel_launch(void* const* d_in, const int* in_sizes, int n_in,
                              void* d_out, int out_size, void* d_ws, size_t ws_size,
                              hipStream_t stream) {
    const float* tokp   = (const float*)d_in[0];  // (8,4096,1024) f32
    const float* expp   = (const float*)d_in[1];  // (64,1024) f32
    const float* cachep = (const float*)d_in[2];  // (64,) f32
    // outputs concatenated flat in return order: int32 indices (32768*2), then f32 weights (32768*2)
    int*   outIdx = (int*)d_out;
    float* outW   = (float*)d_out + (size_t)N_TOK * TOPK;

    dim3 grid(N_TOK / TOK_PER_WG);
    dim3 block(WG_THREADS);
    uzman_kapisi_kernel<<<grid, block, 0, stream>>>(tokp, expp, cachep, outIdx, outW);
}